// RGCN_8564164788309
// MI455X (gfx1250) — compile-verified
//
#include <hip/hip_runtime.h>
#include <hip/hip_bf16.h>

typedef __attribute__((ext_vector_type(2))) float v2f;
typedef __attribute__((ext_vector_type(8))) float v8f;

#define NN 50000   // nodes
#define EE 800000  // edges
#define RR 8       // relations
#define FF 128     // feature dim (IN == H == 128)
#define GRP 4      // relations per aggregation group (2 groups -> accumulator fits L2)

// ---------------------------------------------------------------------------
// Zero the aggregation workspace (S[GRP,NN,FF] followed contiguously by
// cnt[GRP,NN]); float4 grid-stride.
// ---------------------------------------------------------------------------
__global__ __launch_bounds__(256) void rgcn_zero(float4* __restrict__ p, long n4) {
    long i = (long)blockIdx.x * blockDim.x + threadIdx.x;
    long stride = (long)gridDim.x * blockDim.x;
    for (; i < n4; i += stride) p[i] = make_float4(0.f, 0.f, 0.f, 0.f);
}

// ---------------------------------------------------------------------------
// Edge aggregation for one relation group: one wave32 per edge, lane L handles
// features [4L, 4L+4). Edge arrays are streamed non-temporally (keep L2 for
// the accumulator + gathered features). fp32 hardware atomics into S.
// ---------------------------------------------------------------------------
__global__ __launch_bounds__(256) void rgcn_aggregate(
    const int* __restrict__ src, const int* __restrict__ dst,
    const int* __restrict__ etype,
    const float* __restrict__ X,    // [NN, FF]
    float* __restrict__ S,          // [GRP, NN, FF]
    float* __restrict__ cnt,        // [GRP, NN]
    int rbase)
{
    int e    = blockIdx.x * (blockDim.x >> 5) + (threadIdx.x >> 5);
    int lane = threadIdx.x & 31;
    if (e >= EE) return;
    int r = __builtin_nontemporal_load(etype + e) - rbase;
    if ((unsigned)r >= GRP) return;   // whole-wave uniform skip
    int sN = __builtin_nontemporal_load(src + e);
    int dN = __builtin_nontemporal_load(dst + e);
    const float4 v = *(const float4*)(X + (size_t)sN * FF + lane * 4);
    float* o = S + ((size_t)r * NN + dN) * FF + lane * 4;
    unsafeAtomicAdd(o + 0, v.x);
    unsafeAtomicAdd(o + 1, v.y);
    unsafeAtomicAdd(o + 2, v.z);
    unsafeAtomicAdd(o + 3, v.w);
    if (lane == 0) unsafeAtomicAdd(cnt + (size_t)r * NN + dN, 1.0f);
}

// ---------------------------------------------------------------------------
// cnt -> 1 / max(cnt, 1) in place (mean normalization, folded into GEMM A).
// ---------------------------------------------------------------------------
__global__ __launch_bounds__(256) void rgcn_invc(float* __restrict__ c, int n) {
    int i = blockIdx.x * blockDim.x + threadIdx.x;
    if (i < n) c[i] = 1.0f / fmaxf(c[i], 1.0f);
}

// ---------------------------------------------------------------------------
// Pack B operands (root + 8 relation weights) into WMMA order so the GEMM
// fetches each B operand with a single coalesced b64 load:
//   Bp[((seg*32 + k4)*2 + half) * FOUT + col] = { B[ka][col], B[ka+1][col] },
//   ka = 4*k4 + 2*half.  seg 0 = root, segs 1..8 = W_r.
// ---------------------------------------------------------------------------
__global__ __launch_bounds__(256) void rgcn_packB(
    const float* __restrict__ root, const float* __restrict__ W,
    float2* __restrict__ Bp, int fout)
{
    int idx = blockIdx.x * blockDim.x + threadIdx.x;
    int total = (RR + 1) * (FF / 2) * fout;
    if (idx >= total) return;
    int col = idx % fout;
    int t   = idx / fout;
    int k2  = t % (FF / 2);          // row pair (2*k2, 2*k2+1)
    int seg = t / (FF / 2);
    const float* B = (seg == 0) ? root : (W + (size_t)(seg - 1) * FF * fout);
    float2 v = make_float2(B[(size_t)(2 * k2) * fout + col],
                           B[(size_t)(2 * k2 + 1) * fout + col]);
    Bp[(size_t)(seg * (FF / 2) + k2) * fout + col] = v;
}

// ---------------------------------------------------------------------------
// Fused segment GEMM on V_WMMA_F32_16X16X4_F32 (wave32).
// Two accumulation passes per layer:
//   FIRST pass:  segs = [X@root, S_0..S_3];  out = acc + bias
//   second pass: segs = [S_4..S_7]; acc init from out; optional ReLU at end.
// A tiles for all segments are staged once into LDS with the CDNA5 async
// global->LDS path (ASYNCcnt), then consumed via ds_load_b64. Row stride
// padded to FF+4 floats -> conflict-free for 16-row x K-pair access.
// Block = FOUT/16 waves; wave w owns output columns [16w, 16w+16).
// ---------------------------------------------------------------------------
template <int FOUT, int NSEG, bool FIRST, bool RELU>
__global__ __launch_bounds__(256) void rgcn_gemm(
    const float* __restrict__ X,     // node features (seg 0 when FIRST)
    const float* __restrict__ S,     // [GRP, NN, FF] raw aggregates
    const float* __restrict__ invc,  // [GRP, NN]
    const float2* __restrict__ Bp,   // packed B, pre-offset to first segment
    const float* __restrict__ bias,  // [FOUT]
    float* __restrict__ O)           // [NN, FOUT]
{
    __shared__ float tile[NSEG][16][FF + 4];

    const int lane = threadIdx.x & 31;
    const int wave = threadIdx.x >> 5;
    const int half = lane >> 4;      // 0: K pair {k,k+1}; 1: K pair {k+2,k+3}
    const int l16  = lane & 15;
    const int tm   = blockIdx.x * 16;
    const int row  = tm + l16;
    const int col  = wave * 16 + l16;

    // ---- async-stage all A segment tiles (16 x 128 f32 each) into LDS ----
#pragma unroll
    for (int seg = 0; seg < NSEG; ++seg) {
        const float* Asrc;
        if (FIRST) Asrc = (seg == 0) ? X : (S + (size_t)(seg - 1) * NN * FF);
        else       Asrc = S + (size_t)seg * NN * FF;
        for (int f = threadIdx.x; f < (16 * FF) / 4; f += blockDim.x) {
            const int rl = f >> 5;            // FF/4 = 32 float4 per row
            const int c4 = (f & 31) * 4;
            unsigned laddr = (unsigned)(size_t)&tile[seg][rl][c4];
            const float* g = Asrc + (size_t)(tm + rl) * FF + c4;
            asm volatile("global_load_async_to_lds_b128 %0, %1, off"
                         :: "v"(laddr), "v"(g) : "memory");
        }
    }
    asm volatile("s_wait_asynccnt 0x0" ::: "memory");
    __syncthreads();

    // ---- accumulator init ----
    v8f acc;
    if (FIRST) {
#pragma unroll
        for (int i = 0; i < 8; ++i) acc[i] = 0.0f;
    } else {
#pragma unroll
        for (int i = 0; i < 8; ++i)
            acc[i] = O[(size_t)(tm + i + 8 * half) * FOUT + col];
    }

    // ---- WMMA K loop over segments ----
#pragma unroll
    for (int seg = 0; seg < NSEG; ++seg) {
        float scale;
        if (FIRST && seg == 0) scale = 1.0f;
        else {
            const int r = FIRST ? (seg - 1) : seg;
            scale = invc[(size_t)r * NN + row];
        }
#pragma unroll
        for (int k4 = 0; k4 < FF / 4; ++k4) {
            const int ka = k4 * 4 + 2 * half;
            const float2 av = *(const float2*)&tile[seg][l16][ka];
            const float2 bv = Bp[(size_t)((seg * 32 + k4) * 2 + half) * FOUT + col];
            v2f a, b;
            a.x = av.x * scale;
            a.y = av.y * scale;
            b.x = bv.x;
            b.y = bv.y;
            acc = __builtin_amdgcn_wmma_f32_16x16x4_f32(
                false, a, false, b, (short)0, acc, false, false);
        }
    }

    // ---- epilogue ----
#pragma unroll
    for (int i = 0; i < 8; ++i) {
        float v = acc[i];
        if (FIRST) v += bias[col];
        if (RELU)  v = fmaxf(v, 0.0f);
        O[(size_t)(tm + i + 8 * half) * FOUT + col] = v;
    }
}

// ---------------------------------------------------------------------------
// One RGCN layer = pack B + 2 (zero, aggregate-group, invc, gemm) passes.
// ---------------------------------------------------------------------------
template <int FOUT, bool RELU>
static void run_layer(const float* Xf, const float* root, const float* W,
                      const float* bias, float* O,
                      float* S, float* cnt, float2* Bp,
                      const int* srcI, const int* dstI, const int* et,
                      hipStream_t stream)
{
    const int packTot = (RR + 1) * (FF / 2) * FOUT;
    rgcn_packB<<<(packTot + 255) / 256, 256, 0, stream>>>(root, W, Bp, FOUT);

    const long zero4 = ((long)GRP * NN * FF + (long)GRP * NN) / 4;  // S + cnt contiguous
    const int blk = (FOUT / 16) * 32;

    // pass 0: root + relations 0..3, add bias
    rgcn_zero<<<4096, 256, 0, stream>>>((float4*)S, zero4);
    rgcn_aggregate<<<EE / 8, 256, 0, stream>>>(srcI, dstI, et, Xf, S, cnt, 0);
    rgcn_invc<<<(GRP * NN + 255) / 256, 256, 0, stream>>>(cnt, GRP * NN);
    rgcn_gemm<FOUT, 5, true, false><<<NN / 16, blk, 0, stream>>>(Xf, S, cnt, Bp, bias, O);

    // pass 1: relations 4..7, accumulate onto O, optional ReLU
    rgcn_zero<<<4096, 256, 0, stream>>>((float4*)S, zero4);
    rgcn_aggregate<<<EE / 8, 256, 0, stream>>>(srcI, dstI, et, Xf, S, cnt, GRP);
    rgcn_invc<<<(GRP * NN + 255) / 256, 256, 0, stream>>>(cnt, GRP * NN);
    rgcn_gemm<FOUT, 4, false, RELU><<<NN / 16, blk, 0, stream>>>(
        Xf, S, cnt, Bp + (size_t)5 * (FF / 2) * FOUT, bias, O);
}

// ---------------------------------------------------------------------------
// Host-side orchestration (graph-capture safe: kernel launches only).
// ---------------------------------------------------------------------------
extern "C" void kernel_launch(void* const* d_in, const int* in_sizes, int n_in,
                              void* d_out, int out_size, void* d_ws, size_t ws_size,
                              hipStream_t stream) {
    const float* x  = (const float*)d_in[0];
    const int*   ei = (const int*)d_in[1];   // [2, EE]
    const int*   et = (const int*)d_in[2];   // [EE]
    const float* w0 = (const float*)d_in[3];
    const float* r0 = (const float*)d_in[4];
    const float* b0 = (const float*)d_in[5];
    const float* w1 = (const float*)d_in[6];
    const float* r1 = (const float*)d_in[7];
    const float* b1 = (const float*)d_in[8];
    const float* w2 = (const float*)d_in[9];
    const float* r2 = (const float*)d_in[10];
    const float* b2 = (const float*)d_in[11];
    float* out = (float*)d_out;

    // Workspace (floats): h0[NN*FF] h1[NN*FF] S[GRP*NN*FF] cnt[GRP*NN] Bp[9*64*128*2]
    float*  h0  = (float*)d_ws;
    float*  h1  = h0 + (size_t)NN * FF;
    float*  S   = h1 + (size_t)NN * FF;
    float*  cnt = S + (size_t)GRP * NN * FF;
    float2* Bp  = (float2*)(cnt + (size_t)GRP * NN);

    const int* srcI = ei;
    const int* dstI = ei + EE;

    run_layer<128, true >(x,  r0, w0, b0, h0,  S, cnt, Bp, srcI, dstI, et, stream);
    run_layer<128, true >(h0, r1, w1, b1, h1,  S, cnt, Bp, srcI, dstI, et, stream);
    run_layer<16,  false>(h1, r2, w2, b2, out, S, cnt, Bp, srcI, dstI, et, stream);
}